// BNNHAN_65438121722569
// MI455X (gfx1250) — compile-verified
//
#include <hip/hip_runtime.h>

#define NSUB 100000
#define NCH  100000
#define NEDGE 600000
#define DIM  128
#define HEADS 8
#define HEAD_D 16
#define DOUT 4
#define NEG_SLOPE 0.2f

typedef float v2f __attribute__((ext_vector_type(2)));
typedef float v8f __attribute__((ext_vector_type(8)));

// ---------------------------------------------------------------------------
// WMMA f32 GEMM: C[M,128] = A[M,128] @ W[128,128] + bias.  Block tile 64x32,
// 8 waves (4x2), each wave owns a 16x16 tile, K swept in steps of 4 with
// V_WMMA_F32_16X16X4_F32.  LDS padding: A stride 132 (banks 4i, conflict
// free), B stride 48 (lane halves -> disjoint bank ranges).
// ---------------------------------------------------------------------------
__global__ __launch_bounds__(256) void proj_gemm_wmma(
    const float* __restrict__ A, const float* __restrict__ W,
    const float* __restrict__ bias, float* __restrict__ C, int M)
{
  __shared__ float As[64 * 132];
  __shared__ float Bs[128 * 48];
  const int tid = threadIdx.x;
  const int tileM = blockIdx.x * 64;
  const int tileN = blockIdx.y * 32;

#pragma unroll
  for (int i = 0; i < 8; ++i) {              // A tile: 64x128 floats
    int lin = tid + i * 256;                 // 0..2047 float4 slots
    int row = lin >> 5;
    int c4  = (lin & 31) << 2;
    int gr  = tileM + row; if (gr > M - 1) gr = M - 1;
    float4 v = *(const float4*)(A + (size_t)gr * DIM + c4);
    *(float4*)(As + row * 132 + c4) = v;
  }
#pragma unroll
  for (int i = 0; i < 4; ++i) {              // B tile: 128x32 floats
    int lin = tid + i * 256;                 // 0..1023 float4 slots
    int k   = lin >> 3;
    int c4  = (lin & 7) << 2;
    float4 v = *(const float4*)(W + (size_t)k * DIM + tileN + c4);
    *(float4*)(Bs + k * 48 + c4) = v;
  }
  __syncthreads();

  const int lane = tid & 31;
  const int wave = tid >> 5;
  const int wm = wave >> 1, wn = wave & 1;
  const int l16 = lane & 15, lh = lane >> 4;
  const float* ap = As + (wm * 16 + l16) * 132;
  const float* bp = Bs + (wn * 16 + l16);

  v8f acc = {0.f, 0.f, 0.f, 0.f, 0.f, 0.f, 0.f, 0.f};
#pragma unroll
  for (int k0 = 0; k0 < DIM; k0 += 4) {
    v2f a, b;
    a.x = ap[k0 + 2 * lh];            // A 16x4: lane=M + 16*(K>=2), vgpr=K&1
    a.y = ap[k0 + 2 * lh + 1];
    b.x = bp[(k0 + 2 * lh) * 48];     // B 4x16: lane=N + 16*(K>=2), vgpr=K&1
    b.y = bp[(k0 + 2 * lh + 1) * 48];
    acc = __builtin_amdgcn_wmma_f32_16x16x4_f32(false, a, false, b,
                                                (short)0, acc, false, false);
  }

  const int col = tileN + wn * 16 + l16;
  const float bv = bias[col];
#pragma unroll
  for (int r = 0; r < 8; ++r) {       // C 16x16: vgpr r -> M=r (+8 hi half)
    int row = tileM + wm * 16 + r + 8 * lh;
    if (row < M) C[(size_t)row * DIM + col] = acc[r] + bv;
  }
}

// ---------------------------------------------------------------------------
// WMMA GEMM with fused semantic-attention reduction:
//   colsum[z, col] += sum_rows tanh( relu(out_z) @ Wk + bk )
// Never materializes the tanh matrix; each lane atomically adds its partial
// 8-row column sum.
// ---------------------------------------------------------------------------
__global__ __launch_bounds__(256) void tanh_colsum_wmma(
    const float* __restrict__ outCS, const float* __restrict__ outSS,
    const float* __restrict__ Wk, const float* __restrict__ bk,
    float* __restrict__ colsum, int M)
{
  __shared__ float As[64 * 132];
  __shared__ float Bs[128 * 48];
  const float* A = (blockIdx.z == 0) ? outCS : outSS;
  float* cs = colsum + blockIdx.z * DIM;
  const int tid = threadIdx.x;
  const int tileM = blockIdx.x * 64;
  const int tileN = blockIdx.y * 32;

#pragma unroll
  for (int i = 0; i < 8; ++i) {
    int lin = tid + i * 256;
    int row = lin >> 5;
    int c4  = (lin & 31) << 2;
    int gr  = tileM + row; if (gr > M - 1) gr = M - 1;
    float4 v = *(const float4*)(A + (size_t)gr * DIM + c4);
    v.x = fmaxf(v.x, 0.f); v.y = fmaxf(v.y, 0.f);   // relu fused into load
    v.z = fmaxf(v.z, 0.f); v.w = fmaxf(v.w, 0.f);
    *(float4*)(As + row * 132 + c4) = v;
  }
#pragma unroll
  for (int i = 0; i < 4; ++i) {
    int lin = tid + i * 256;
    int k   = lin >> 3;
    int c4  = (lin & 7) << 2;
    float4 v = *(const float4*)(Wk + (size_t)k * DIM + tileN + c4);
    *(float4*)(Bs + k * 48 + c4) = v;
  }
  __syncthreads();

  const int lane = tid & 31;
  const int wave = tid >> 5;
  const int wm = wave >> 1, wn = wave & 1;
  const int l16 = lane & 15, lh = lane >> 4;
  const float* ap = As + (wm * 16 + l16) * 132;
  const float* bp = Bs + (wn * 16 + l16);

  v8f acc = {0.f, 0.f, 0.f, 0.f, 0.f, 0.f, 0.f, 0.f};
#pragma unroll
  for (int k0 = 0; k0 < DIM; k0 += 4) {
    v2f a, b;
    a.x = ap[k0 + 2 * lh];
    a.y = ap[k0 + 2 * lh + 1];
    b.x = bp[(k0 + 2 * lh) * 48];
    b.y = bp[(k0 + 2 * lh + 1) * 48];
    acc = __builtin_amdgcn_wmma_f32_16x16x4_f32(false, a, false, b,
                                                (short)0, acc, false, false);
  }

  const int col = tileN + wn * 16 + l16;
  const float bkv = bk[col];
  float part = 0.f;
#pragma unroll
  for (int r = 0; r < 8; ++r) {
    int row = tileM + wm * 16 + r + 8 * lh;
    part += (row < M) ? tanhf(acc[r] + bkv) : 0.f;
  }
  atomicAdd(cs + col, part);
}

// --------------------------------------------------------------------------
// a[n,h] = sum_d h[n, h*16+d] * att[h, d]
// --------------------------------------------------------------------------
__global__ __launch_bounds__(256) void node_att_k(
    const float* __restrict__ h, const float* __restrict__ att,
    float* __restrict__ out, int N)
{
  int idx = blockIdx.x * 256 + threadIdx.x;
  if (idx >= N * HEADS) return;
  int n = idx >> 3, hh = idx & 7;
  const float* hp = h + (size_t)n * DIM + hh * HEAD_D;
  const float* ap = att + hh * HEAD_D;
  float acc = 0.f;
#pragma unroll
  for (int d = 0; d < HEAD_D; d += 4) {
    float4 hv = *(const float4*)(hp + d);
    float4 av = *(const float4*)(ap + d);
    acc += hv.x * av.x + hv.y * av.y + hv.z * av.z + hv.w * av.w;
  }
  out[idx] = acc;
}

// --------------------------------------------------------------------------
// Edge pass 1: ex = exp(leaky_relu(a_s[src]+a_d[dst])); den[dst] += ex.
// (Max-shift omitted: softmax is shift-invariant and logits are O(0.2).)
// --------------------------------------------------------------------------
__global__ __launch_bounds__(256) void edge_pass1_k(
    const int* __restrict__ edge, const float* __restrict__ a_s,
    const float* __restrict__ a_d, float* __restrict__ exbuf,
    float* __restrict__ den)
{
  int idx = blockIdx.x * 256 + threadIdx.x;
  if (idx >= NEDGE * HEADS) return;
  int e = idx >> 3, h = idx & 7;
  int s = edge[e], d = edge[NEDGE + e];
  float al = a_s[s * 8 + h] + a_d[d * 8 + h];
  al = (al >= 0.f) ? al : al * NEG_SLOPE;
  float ex = __expf(al);
  exbuf[idx] = ex;
  atomicAdd(den + d * 8 + h, ex);
}

// --------------------------------------------------------------------------
// Edge pass 2: out[dst, c] += h_src[src, c] * ex/(den[dst]+eps).  128 threads
// per edge -> coalesced row gather, f32 global atomic scatter.
// --------------------------------------------------------------------------
__global__ __launch_bounds__(256) void edge_pass2_k(
    const int* __restrict__ edge, const float* __restrict__ hsrc,
    const float* __restrict__ exbuf, const float* __restrict__ den,
    float* __restrict__ out)
{
  long long idx = (long long)blockIdx.x * 256 + threadIdx.x;
  if (idx >= (long long)NEDGE * DIM) return;
  int e = (int)(idx >> 7);
  int c = (int)(idx & 127);
  int h = c >> 4;
  int s = edge[e], d = edge[NEDGE + e];
  float w = exbuf[e * 8 + h] / (den[d * 8 + h] + 1e-16f);
  atomicAdd(out + (size_t)d * DIM + c, hsrc[(size_t)s * DIM + c] * w);
}

// --------------------------------------------------------------------------
// Semantic softmax over the 2 edge types from the fused column sums.
// --------------------------------------------------------------------------
__global__ void semantic_score_k(const float* __restrict__ colsum,
                                 const float* __restrict__ q,
                                 float* __restrict__ attn)
{
  __shared__ float r0[128], r1[128];
  int f = threadIdx.x;
  float qf = q[f];
  const float invN = 1.0f / (float)NSUB;
  r0[f] = qf * colsum[f] * invN;
  r1[f] = qf * colsum[DIM + f] * invN;
  __syncthreads();
  for (int s = 64; s > 0; s >>= 1) {
    if (f < s) { r0[f] += r0[f + s]; r1[f] += r1[f + s]; }
    __syncthreads();
  }
  if (f == 0) {
    float s0 = r0[0], s1 = r1[0];
    float mx = fmaxf(s0, s1);
    float e0 = __expf(s0 - mx), e1 = __expf(s1 - mx);
    float inv = 1.f / (e0 + e1);
    attn[0] = e0 * inv;
    attn[1] = e1 * inv;
  }
}

// --------------------------------------------------------------------------
// y[n,:] = (attn0*relu(out_cs[n]) + attn1*relu(out_ss[n])) @ Wo + bo
// --------------------------------------------------------------------------
__global__ __launch_bounds__(256) void final_k(
    const float* __restrict__ outCS, const float* __restrict__ outSS,
    const float* __restrict__ Wo, const float* __restrict__ bo,
    const float* __restrict__ attn, float* __restrict__ y)
{
  __shared__ float WoS[DIM * DOUT];
  int tid = threadIdx.x;
  ((float2*)WoS)[tid] = ((const float2*)Wo)[tid];   // 256 thr x float2 = 512 f
  __syncthreads();
  float a0 = attn[0], a1 = attn[1];
  int n = blockIdx.x * 256 + tid;
  if (n >= NSUB) return;
  const float* pc = outCS + (size_t)n * DIM;
  const float* ps = outSS + (size_t)n * DIM;
  float acc0 = 0.f, acc1 = 0.f, acc2 = 0.f, acc3 = 0.f;
#pragma unroll 4
  for (int k = 0; k < DIM; ++k) {
    float x = a0 * fmaxf(pc[k], 0.f) + a1 * fmaxf(ps[k], 0.f);
    acc0 += x * WoS[k * 4 + 0];
    acc1 += x * WoS[k * 4 + 1];
    acc2 += x * WoS[k * 4 + 2];
    acc3 += x * WoS[k * 4 + 3];
  }
  float4 r = {acc0 + bo[0], acc1 + bo[1], acc2 + bo[2], acc3 + bo[3]};
  *(float4*)(y + (size_t)n * 4) = r;
}

extern "C" void kernel_launch(void* const* d_in, const int* in_sizes, int n_in,
                              void* d_out, int out_size, void* d_ws, size_t ws_size,
                              hipStream_t stream) {
  (void)in_sizes; (void)n_in; (void)out_size; (void)ws_size;
  const float* x_sub   = (const float*)d_in[0];
  const float* x_ch    = (const float*)d_in[1];
  const int*   e_cs    = (const int*)d_in[2];
  const int*   e_ss    = (const int*)d_in[3];
  const float* Wl_sub  = (const float*)d_in[4];
  const float* bl_sub  = (const float*)d_in[5];
  const float* Wl_ch   = (const float*)d_in[6];
  const float* bl_ch   = (const float*)d_in[7];
  const float* att_s_cs = (const float*)d_in[8];
  const float* att_d_cs = (const float*)d_in[9];
  const float* att_s_ss = (const float*)d_in[10];
  const float* att_d_ss = (const float*)d_in[11];
  const float* Wk = (const float*)d_in[12];
  const float* bk = (const float*)d_in[13];
  const float* q  = (const float*)d_in[14];
  const float* Wo = (const float*)d_in[15];
  const float* bo = (const float*)d_in[16];
  float* y = (float*)d_out;

  // workspace carve-up (floats)
  float* p = (float*)d_ws;
  float* h_sub  = p; p += (size_t)NSUB * DIM;
  float* h_ch   = p; p += (size_t)NCH * DIM;
  float* a_s_cs = p; p += (size_t)NCH * HEADS;
  float* a_d_cs = p; p += (size_t)NSUB * HEADS;
  float* a_s_ss = p; p += (size_t)NSUB * HEADS;
  float* a_d_ss = p; p += (size_t)NSUB * HEADS;
  float* ex_cs  = p; p += (size_t)NEDGE * HEADS;
  float* ex_ss  = p; p += (size_t)NEDGE * HEADS;
  float* den_cs = p; p += (size_t)NSUB * HEADS;
  float* den_ss = p; p += (size_t)NSUB * HEADS;
  float* out_cs = p; p += (size_t)NSUB * DIM;
  float* out_ss = p; p += (size_t)NSUB * DIM;
  float* colsum = p; p += 2 * DIM;
  float* attn   = p; p += 2;

  // zero accumulators (den_cs|den_ss, out_cs|out_ss, colsum|attn contiguous)
  hipMemsetAsync(den_cs, 0, sizeof(float) * (size_t)NSUB * HEADS * 2, stream);
  hipMemsetAsync(out_cs, 0, sizeof(float) * (size_t)NSUB * DIM * 2, stream);
  hipMemsetAsync(colsum, 0, sizeof(float) * (2 * DIM + 2), stream);

  // 1) node projections (WMMA f32)
  dim3 gp((NSUB + 63) / 64, DIM / 32);
  proj_gemm_wmma<<<gp, 256, 0, stream>>>(x_sub, Wl_sub, bl_sub, h_sub, NSUB);
  proj_gemm_wmma<<<gp, 256, 0, stream>>>(x_ch,  Wl_ch,  bl_ch,  h_ch,  NCH);

  // 2) per-node attention logits
  int nab = (NSUB * HEADS + 255) / 256;
  node_att_k<<<nab, 256, 0, stream>>>(h_ch,  att_s_cs, a_s_cs, NCH);
  node_att_k<<<nab, 256, 0, stream>>>(h_sub, att_d_cs, a_d_cs, NSUB);
  node_att_k<<<nab, 256, 0, stream>>>(h_sub, att_s_ss, a_s_ss, NSUB);
  node_att_k<<<nab, 256, 0, stream>>>(h_sub, att_d_ss, a_d_ss, NSUB);

  // 3) segment softmax numerators + denominators
  int ep1 = (NEDGE * HEADS + 255) / 256;
  edge_pass1_k<<<ep1, 256, 0, stream>>>(e_cs, a_s_cs, a_d_cs, ex_cs, den_cs);
  edge_pass1_k<<<ep1, 256, 0, stream>>>(e_ss, a_s_ss, a_d_ss, ex_ss, den_ss);

  // 4) weighted message scatter-add
  long long tot2 = (long long)NEDGE * DIM;
  int ep2 = (int)((tot2 + 255) / 256);
  edge_pass2_k<<<ep2, 256, 0, stream>>>(e_cs, h_ch,  ex_cs, den_cs, out_cs);
  edge_pass2_k<<<ep2, 256, 0, stream>>>(e_ss, h_sub, ex_ss, den_ss, out_ss);

  // 5) fused semantic GEMM + column-mean reduction (WMMA f32)
  dim3 gt((NSUB + 63) / 64, DIM / 32, 2);
  tanh_colsum_wmma<<<gt, 256, 0, stream>>>(out_cs, out_ss, Wk, bk, colsum, NSUB);

  // 6) semantic softmax (2 scalars)
  semantic_score_k<<<1, 128, 0, stream>>>(colsum, q, attn);

  // 7) final classifier
  final_k<<<(NSUB + 255) / 256, 256, 0, stream>>>(out_cs, out_ss, Wo, bo, attn, y);
}